// PostProcessor_50233937494363
// MI455X (gfx1250) — compile-verified
//
#include <hip/hip_runtime.h>
#include <stdint.h>

#define NB 16
#define NP 2048
#define SCORE_THRESH 0.05f
#define NMS_THRESH 0.5f
#define BBOX_CLIP 4.135166556742356f
#define IMG_W 1333.0f
#define IMG_H 800.0f

typedef float v2f __attribute__((ext_vector_type(2)));
typedef float v8f __attribute__((ext_vector_type(8)));
typedef unsigned int u32x4 __attribute__((ext_vector_type(4)));
typedef int i32x4 __attribute__((ext_vector_type(4)));
typedef int i32x8 __attribute__((ext_vector_type(8)));

// ---------------- Kernel 1: decode + score + bitonic sort (per image) -------
__global__ __launch_bounds__(1024) void k1_decode_sort(
    const float* __restrict__ logits, const float* __restrict__ reg,
    const float* __restrict__ props, float* __restrict__ sbox,
    float* __restrict__ skey)
{
  __shared__ float key[NP];            // 8 KB
  __shared__ unsigned short sidx[NP];  // 4 KB
  __shared__ float box4[NP * 4];       // 32 KB
  const int b = blockIdx.x;
  const int tid = threadIdx.x;

  for (int p = tid; p < NP; p += 1024) {
    float l0 = logits[((size_t)b * NP + p) * 2 + 0];
    float l1 = logits[((size_t)b * NP + p) * 2 + 1];
    float score = 1.0f / (1.0f + __expf(l0 - l1));   // softmax class-1
    const float* c = &reg[((size_t)b * NP + p) * 8];
    float dx = c[4] * 0.1f;
    float dy = c[5] * 0.1f;
    float dw = fminf(c[6] * 0.2f, BBOX_CLIP);
    float dh = fminf(c[7] * 0.2f, BBOX_CLIP);
    const float* pb = &props[((size_t)b * NP + p) * 4];
    float w = pb[2] - pb[0] + 1.0f;
    float h = pb[3] - pb[1] + 1.0f;
    float cx = pb[0] + 0.5f * w, cy = pb[1] + 0.5f * h;
    float pcx = dx * w + cx, pcy = dy * h + cy;
    float pw = __expf(dw) * w, ph = __expf(dh) * h;
    float nx1 = fminf(fmaxf(pcx - 0.5f * pw, 0.f), IMG_W - 1.0f);
    float ny1 = fminf(fmaxf(pcy - 0.5f * ph, 0.f), IMG_H - 1.0f);
    float nx2 = fminf(fmaxf(pcx + 0.5f * pw - 1.0f, 0.f), IMG_W - 1.0f);
    float ny2 = fminf(fmaxf(pcy + 0.5f * ph - 1.0f, 0.f), IMG_H - 1.0f);
    box4[p * 4 + 0] = nx1; box4[p * 4 + 1] = ny1;
    box4[p * 4 + 2] = nx2; box4[p * 4 + 3] = ny2;
    key[p] = (score > SCORE_THRESH) ? score : -1.0f; // invalid sinks to the end
    sidx[p] = (unsigned short)p;
  }
  __syncthreads();

  // Bitonic sort, descending by key
  for (unsigned k = 2; k <= NP; k <<= 1) {
    for (unsigned j = k >> 1; j > 0; j >>= 1) {
      for (unsigned t = tid; t < NP; t += 1024) {
        unsigned ixj = t ^ j;
        if (ixj > t) {
          bool up = ((t & k) == 0);
          float ka = key[t], kb = key[ixj];
          bool sw = up ? (ka < kb) : (ka > kb);
          if (sw) {
            key[t] = kb; key[ixj] = ka;
            unsigned short ia = sidx[t]; sidx[t] = sidx[ixj]; sidx[ixj] = ia;
          }
        }
      }
      __syncthreads();
    }
  }

  for (int p = tid; p < NP; p += 1024) {
    int o = sidx[p];
    float* d = &sbox[((size_t)b * NP + p) * 4];
    d[0] = box4[o * 4 + 0]; d[1] = box4[o * 4 + 1];
    d[2] = box4[o * 4 + 2]; d[3] = box4[o * 4 + 3];
    skey[(size_t)b * NP + p] = key[p];
  }
}

// ---------------- Kernel 2: suppression bitmask via WMMA tiles + TDM -------
__global__ __launch_bounds__(256) void k2_mask(
    const float* __restrict__ sbox, uint32_t* __restrict__ mask)
{
  __shared__ float bxs[NP * 4];        // 32 KB sorted boxes (TDM-loaded)
  __shared__ float area[NP];           // 8 KB
  __shared__ unsigned stage[8 * 16];   // per-wave row-word staging
  const int b = blockIdx.x >> 3;
  const int chunk = blockIdx.x & 7;    // 8 chunks x 256 rows
  const int tid = threadIdx.x;
  const int lane = tid & 31, wave = tid >> 5;
  const int n = lane & 15, half = lane >> 4;

  if (wave == 0) {
    // Tensor Data Mover: 1-row tensor of 8192 f32 (32 KB), global -> LDS
    uint64_t ga = (uint64_t)(uintptr_t)(sbox + (size_t)b * NP * 4);
    uint32_t lo = (uint32_t)(uintptr_t)(&bxs[0]);   // LDS byte offset
    u32x4 g0;
    g0.x = 1u;                                        // count=1, user mode
    g0.y = lo;                                        // lds_addr
    g0.z = (uint32_t)ga;                              // global_addr[31:0]
    g0.w = ((uint32_t)(ga >> 32) & 0x1FFFFFFu) | (2u << 30); // addr[56:32]|type=2
    i32x8 g1;
    g1[0] = 0x00020000;                  // wg_mask=0, data_size=2 (4B)
    g1[1] = (int)(((NP * 4) & 0xFFFF) << 16);          // tensor_dim0 lo16
    g1[2] = (int)(1u << 16);                           // dim0 hi16=0 | tensor_dim1=1
    g1[3] = (int)((unsigned)(NP * 4) << 16);           // tile_dim0 = 8192
    g1[4] = 1;                           // tile_dim1=1, tile_dim2=0
    g1[5] = NP * 4;                      // tensor_dim0_stride lo32
    g1[6] = 0; g1[7] = 0;
    i32x4 gz4 = {0, 0, 0, 0};
    i32x8 gz8 = {0, 0, 0, 0, 0, 0, 0, 0};
    __builtin_amdgcn_tensor_load_to_lds(g0, g1, gz4, gz4, gz8, 0);
    __builtin_amdgcn_s_wait_tensorcnt(0);
  }
  if (tid < 128) stage[tid] = 0;
  __syncthreads();

  for (int p = tid; p < NP; p += 256) {
    float x1 = bxs[p * 4 + 0], y1 = bxs[p * 4 + 1];
    float x2 = bxs[p * 4 + 2], y2 = bxs[p * 4 + 3];
    area[p] = (x2 - x1) * (y2 - y1);
  }
  __syncthreads();

  const unsigned slot = wave * 16;
  const float zsel = half ? 0.0f : 1.0f;   // lane-half select as a multiplier
  for (int rt = 0; rt < 2; ++rt) {
    const int rowbase = chunk * 256 + (wave * 2 + rt) * 16;
    // A (16x4 f32): row i = [area_i, 1, 0, 0]; lanes 16-31 carry K=2,3 = 0
    float an = area[rowbase + n];
    v2f a;
    a.x = an * zsel;
    a.y = zsel;
    // Preload this lane's 8 row boxes (invariant across all column tiles)
    float rx1[8], ry1[8], rx2[8], ry2[8];
#pragma unroll
    for (int vg = 0; vg < 8; ++vg) {
      int i = rowbase + vg + 8 * half;
      rx1[vg] = bxs[i * 4 + 0]; ry1[vg] = bxs[i * 4 + 1];
      rx2[vg] = bxs[i * 4 + 2]; ry2[vg] = bxs[i * 4 + 3];
    }
    for (int w = 0; w < 64; ++w) {
      for (int sub = 0; sub < 2; ++sub) {
        const int colbase = (w * 2 + sub) * 16;
        const int j = colbase + n;
        // B (4x16 f32): col j = [1, area_j, 0, 0]^T (unconditional load + select)
        float aj = area[j];
        v2f bb;
        bb.x = zsel;
        bb.y = aj * zsel;
        v8f d = {0, 0, 0, 0, 0, 0, 0, 0};
        // D[i][j] = area_i + area_j over the 16x16 tile
        d = __builtin_amdgcn_wmma_f32_16x16x4_f32(
            false, a, false, bb, (short)0, d, false, false);
        float jx1 = bxs[j * 4 + 0], jy1 = bxs[j * 4 + 1];
        float jx2 = bxs[j * 4 + 2], jy2 = bxs[j * 4 + 3];
        unsigned bit = 1u << (j & 31);
#pragma unroll
        for (int vg = 0; vg < 8; ++vg) {
          int i = rowbase + vg + 8 * half;   // C/D layout: M = vg + 8*half
          float ltx = fmaxf(rx1[vg], jx1), lty = fmaxf(ry1[vg], jy1);
          float rbx = fminf(rx2[vg], jx2), rby = fminf(ry2[vg], jy2);
          float wx = fmaxf(rbx - ltx, 0.0f), wy = fmaxf(rby - lty, 0.0f);
          float inter = wx * wy;
          float denom = d[vg] - inter + 1e-9f;
          // iou > NMS_THRESH  <=>  inter > T*denom when denom >= 0;
          // denom < 0 => reference iou <= 0 => never suppress.
          if ((inter > NMS_THRESH * denom) && (denom >= 0.0f) && (j > i)) {
            atomicOr(&stage[slot + vg + 8 * half], bit);
          }
        }
      }
      __threadfence_block();
      if (lane < 16) {
        int i = rowbase + lane;
        mask[(((size_t)b * NP) + i) * 64 + w] = stage[slot + lane];
        stage[slot + lane] = 0;
      }
      __threadfence_block();
    }
  }
}

// ---------------- Kernel 3: sequential greedy NMS scan (one wave/image) ----
__global__ __launch_bounds__(32) void k3_scan(
    const float* __restrict__ skey, const uint32_t* __restrict__ mask,
    uint32_t* __restrict__ keep)
{
  const int b = blockIdx.x;
  const int lane = threadIdx.x;
  unsigned keepA = 0, keepB = 0;   // lane L owns words L and L+32 (2048 bits)
  for (int t = 0; t < 32; ++t) {
    keepA |= (skey[(size_t)b * NP + lane * 32 + t] > SCORE_THRESH ? 1u : 0u) << t;
    keepB |= (skey[(size_t)b * NP + (lane + 32) * 32 + t] > SCORE_THRESH ? 1u : 0u) << t;
  }
  for (int i = 0; i < NP; ++i) {
    int wi = i >> 5;
    unsigned word = (wi < 32) ? (unsigned)__shfl((int)keepA, wi, 32)
                              : (unsigned)__shfl((int)keepB, wi - 32, 32);
    if ((word >> (i & 31)) & 1u) {     // proposal i survives -> it suppresses
      const uint32_t* mr = &mask[(((size_t)b * NP) + i) * 64];
      keepA &= ~mr[lane];
      keepB &= ~mr[lane + 32];
    }
  }
  keep[b * 64 + lane] = keepA;
  keep[b * 64 + 32 + lane] = keepB;
}

// ---------------- Kernel 4: masked writeout -------------------------------
__global__ __launch_bounds__(256) void k4_out(
    const float* __restrict__ sbox, const float* __restrict__ skey,
    const uint32_t* __restrict__ keep, float* __restrict__ out)
{
  int t = blockIdx.x * blockDim.x + threadIdx.x;
  if (t >= NB * NP) return;
  int b = t >> 11, p = t & (NP - 1);
  unsigned k = (keep[b * 64 + (p >> 5)] >> (p & 31)) & 1u;
  float m = k ? 1.0f : 0.0f;
  float* o = &out[(size_t)t * 5];
  const float* s = &sbox[(size_t)t * 4];
  o[0] = s[0] * m; o[1] = s[1] * m; o[2] = s[2] * m; o[3] = s[3] * m;
  o[4] = skey[t] * m;
}

extern "C" void kernel_launch(void* const* d_in, const int* in_sizes, int n_in,
                              void* d_out, int out_size, void* d_ws, size_t ws_size,
                              hipStream_t stream)
{
  const float* logits = (const float*)d_in[0];
  const float* reg    = (const float*)d_in[1];
  const float* props  = (const float*)d_in[2];
  char* ws = (char*)d_ws;
  float*    sbox = (float*)ws;                               // 512 KB
  float*    skey = (float*)(ws + 512 * 1024);                // 128 KB
  uint32_t* mask = (uint32_t*)(ws + 640 * 1024);             // 8 MB
  uint32_t* keep = (uint32_t*)(ws + 640 * 1024 + 8 * 1024 * 1024); // 4 KB

  k1_decode_sort<<<NB, 1024, 0, stream>>>(logits, reg, props, sbox, skey);
  k2_mask<<<NB * 8, 256, 0, stream>>>(sbox, mask);
  k3_scan<<<NB, 32, 0, stream>>>(skey, mask, keep);
  k4_out<<<(NB * NP + 255) / 256, 256, 0, stream>>>(sbox, skey, keep, (float*)d_out);
}